// AttentionLayer_3238405341397
// MI455X (gfx1250) — compile-verified
//
#include <hip/hip_runtime.h>
#include <math.h>

// MI455X fused attention layer, fp32 WMMA (16x16x4) end-to-end.
// Roofline: ~34 GFLOP vs ~130MB traffic @23.3TB/s -> memory bound, so we keep
// full fp32 precision (matches reference bit-for-bit semantics) and fuse all
// phases so scores/context never round-trip to HBM (E and W stay L2-resident).

typedef __attribute__((ext_vector_type(2))) float v2f;
typedef __attribute__((ext_vector_type(8))) float v8f;

#define NB   32
#define SD   512
#define SE   512
#define DD   512
#define STR  516   // LDS row stride in floats: +4 dwords -> conflict-free b64 reads

__launch_bounds__(256, 1)
__global__ void attn_fused_kernel(const float* __restrict__ Q,     // (B,SD,D)
                                  const float* __restrict__ E,     // (B,SE,D)
                                  const float* __restrict__ mask,  // (B,SD,1)
                                  const float* __restrict__ W,     // (D, 2D)
                                  const float* __restrict__ bias,  // (D)
                                  float* __restrict__ outMasked,   // (B,SD,D)
                                  float* __restrict__ outWeights)  // (B,SD,SE)
{
    __shared__ float sQ[16 * STR];      // Q tile 16x512
    __shared__ float sS[16 * STR];      // scores -> probs -> context
    __shared__ float sRed[16 * 16];     // softmax reductions
    __shared__ float sMask[16];

    const int b    = blockIdx.y;
    const int q0   = blockIdx.x * 16;
    const int tid  = threadIdx.x;
    const int lane = tid & 31;
    const int wave = tid >> 5;          // 0..7
    const int half = lane >> 4;         // 0 / 1 (K-halves 0,1 vs 2,3)
    const int l16  = lane & 15;

    const float* Qb = Q + ((size_t)b * SD + q0) * DD;
    const float* Eb = E + (size_t)b * SE * DD;

    // ---- Phase 0: stage Q tile + mask into LDS (coalesced) ----
    for (int i = tid; i < 16 * DD; i += 256) {
        int r = i >> 9;         // /512
        int c = i & 511;
        sQ[r * STR + c] = Qb[r * DD + c];
    }
    if (tid < 16) sMask[tid] = mask[(size_t)b * SD + q0 + tid];
    __syncthreads();

    const float scale = 0.044194173824159216f;   // 1/sqrt(512)

    // ---- Phase 1: scores = Q * E^T, each wave owns 4 k-tiles of 16 ----
    {
        const float* Arow = sQ + l16 * STR + 2 * half;   // A[M=l16][d + 2*half + {0,1}]
        for (int t = 0; t < 4; ++t) {
            const int k0 = (wave * 4 + t) * 16;
            const float* Brow = Eb + (size_t)(k0 + l16) * DD + 2 * half; // B[kk][n]=E[k0+n][d+kk]
            v8f acc = {};
            for (int d = 0; d < DD; d += 4) {
                v2f a;  a.x = Arow[d];  a.y = Arow[d + 1];
                v2f bf; bf.x = Brow[d]; bf.y = Brow[d + 1];
                acc = __builtin_amdgcn_wmma_f32_16x16x4_f32(
                    false, a, false, bf, (short)0, acc, false, false);
            }
#pragma unroll
            for (int i = 0; i < 8; ++i)      // C/D: M = i + 8*half, N = l16
                sS[(i + 8 * half) * STR + k0 + l16] = acc[i] * scale;
        }
    }
    __syncthreads();

    // ---- Phase 2: row softmax (16 rows x 512, 16 threads per row) ----
    {
        const int row = tid & 15;
        const int sub = tid >> 4;        // 0..15, owns 32 columns
        float* srow = sS + row * STR;
        float m = -INFINITY;
        for (int c = sub * 32; c < sub * 32 + 32; ++c) m = fmaxf(m, srow[c]);
        sRed[row * 16 + sub] = m;
        __syncthreads();
        float mall = -INFINITY;
#pragma unroll
        for (int j = 0; j < 16; ++j) mall = fmaxf(mall, sRed[row * 16 + j]);
        __syncthreads();
        float s = 0.f;
        for (int c = sub * 32; c < sub * 32 + 32; ++c) {
            float p = __expf(srow[c] - mall);
            srow[c] = p;
            s += p;
        }
        sRed[row * 16 + sub] = s;
        __syncthreads();
        float stot = 0.f;
#pragma unroll
        for (int j = 0; j < 16; ++j) stot += sRed[row * 16 + j];
        const float inv = 1.0f / stot;
        float* gw = outWeights + ((size_t)b * SD + q0 + row) * SE;
        for (int c = sub * 32; c < sub * 32 + 32; ++c) {
            float p = srow[c] * inv;
            srow[c] = p;       // normalized probs kept for phase 3
            gw[c]   = p;       // attention_weights output
        }
    }
    __syncthreads();

    // ---- Phase 3: ctx = P * E, each wave owns 4 n-tiles (64 cols) ----
    v8f ctx[4];
    {
        const float* Arow = sS + l16 * STR + 2 * half;   // A[M=l16][k + 2*half + {0,1}]
        for (int t = 0; t < 4; ++t) {
            const int n0 = wave * 64 + t * 16;
            const float* Bcol = Eb + n0 + l16;           // B[kk][n]=E[k+kk][n0+n]
            v8f acc = {};
            for (int k = 0; k < SE; k += 4) {
                v2f a;  a.x = Arow[k]; a.y = Arow[k + 1];
                v2f bf;
                bf.x = Bcol[(size_t)(k + 2 * half) * DD];        // coalesced 16-lane rows
                bf.y = Bcol[(size_t)(k + 2 * half + 1) * DD];
                acc = __builtin_amdgcn_wmma_f32_16x16x4_f32(
                    false, a, false, bf, (short)0, acc, false, false);
            }
            ctx[t] = acc;
        }
    }
    __syncthreads();   // everyone done reading probs from sS
#pragma unroll
    for (int t = 0; t < 4; ++t) {
        const int n0 = wave * 64 + t * 16;
#pragma unroll
        for (int i = 0; i < 8; ++i)
            sS[(i + 8 * half) * STR + n0 + l16] = ctx[t][i];   // sS now holds ctx
    }
    __syncthreads();

    // ---- Phase 4: out = tanh([Q|ctx] @ W^T + b) * mask ----
    {
        const float* AQ = sQ + l16 * STR + 2 * half;
        const float* AC = sS + l16 * STR + 2 * half;
        for (int t = 0; t < 4; ++t) {
            const int n0 = wave * 64 + t * 16;
            // B[kk][n] = W[n0+n][k+kk]; W row = 2*DD floats
            const float* Wr = W + (size_t)(n0 + l16) * (2 * DD) + 2 * half;
            v8f acc = {};
            for (int k = 0; k < DD; k += 4) {              // k in [0,512): cat = Q
                v2f a;  a.x = AQ[k]; a.y = AQ[k + 1];
                v2f bf; bf.x = Wr[k]; bf.y = Wr[k + 1];
                acc = __builtin_amdgcn_wmma_f32_16x16x4_f32(
                    false, a, false, bf, (short)0, acc, false, false);
            }
            const float* Wr2 = Wr + DD;                    // k in [512,1024): cat = ctx
            for (int k = 0; k < DD; k += 4) {
                v2f a;  a.x = AC[k]; a.y = AC[k + 1];
                v2f bf; bf.x = Wr2[k]; bf.y = Wr2[k + 1];
                acc = __builtin_amdgcn_wmma_f32_16x16x4_f32(
                    false, a, false, bf, (short)0, acc, false, false);
            }
            const float bn = bias[n0 + l16];
            float* orow = outMasked + ((size_t)b * SD + q0) * DD + n0 + l16;
#pragma unroll
            for (int i = 0; i < 8; ++i) {
                const int M = i + 8 * half;
                orow[(size_t)M * DD] = tanhf(acc[i] + bn) * sMask[M];
            }
        }
    }
}

extern "C" void kernel_launch(void* const* d_in, const int* in_sizes, int n_in,
                              void* d_out, int out_size, void* d_ws, size_t ws_size,
                              hipStream_t stream) {
    (void)in_sizes; (void)n_in; (void)out_size; (void)d_ws; (void)ws_size;
    const float* Q    = (const float*)d_in[0];   // padded_seqs (B,SD,D)
    const float* E    = (const float*)d_in[1];   // encoder_padded_seqs (B,SE,D)
    const float* mask = (const float*)d_in[2];   // decoder_mask (B,SD,1)
    const float* W    = (const float*)d_in[3];   // (D, 2D)
    const float* bias = (const float*)d_in[4];   // (D)
    float* outMasked  = (float*)d_out;                                   // (B,SD,D)
    float* outWeights = (float*)d_out + (size_t)NB * SD * SE;            // (B,SD,SE)

    dim3 grid(SD / 16, NB);   // 32 x 32 workgroups
    attn_fused_kernel<<<grid, 256, 0, stream>>>(Q, E, mask, W, bias,
                                                outMasked, outWeights);
}